// ESN_19344532701985
// MI455X (gfx1250) — compile-verified
//
#include <hip/hip_runtime.h>
#include <math.h>

// Problem constants from the reference
#define T_STEPS 8192
#define IN_DIM  256
#define OUT_DIM 256
#define RES_DIM 4096

// Persistent-scan configuration
#define NB      256                 // persistent workgroups (one pred row each)
#define BS      256                 // threads per workgroup (8 waves, wave32)
#define ROWS_PB (RES_DIM / NB)      // 16 W_res rows per block
#define LPR     16                  // lanes cooperating per row

typedef float v2f __attribute__((ext_vector_type(2)));
typedef float v8f __attribute__((ext_vector_type(8)));

// ---------------------------------------------------------------------------
// Phase 0: init — zero barrier counter, load state0 into state buffer A.
// ---------------------------------------------------------------------------
__global__ __launch_bounds__(256)
void esn_init(const float* __restrict__ state0,
              float* __restrict__ sA, float* __restrict__ sB,
              unsigned int* __restrict__ cnt) {
    int i = blockIdx.x * blockDim.x + threadIdx.x;
    if (i < RES_DIM) { sA[i] = state0[i]; sB[i] = 0.0f; }
    if (i == 0) *cnt = 0u;
}

// ---------------------------------------------------------------------------
// Phase 1: U = X @ W_in^T  via fp32 WMMA (16x16x4).  (codegen-verified)
// ---------------------------------------------------------------------------
__global__ __launch_bounds__(128)
void esn_win_gemm_wmma(const float* __restrict__ X,
                       const float* __restrict__ W_in,
                       float* __restrict__ U) {
    const int wave = blockIdx.x * (blockDim.x >> 5) + (threadIdx.x >> 5);
    const int lane = threadIdx.x & 31;
    const int half = lane >> 4;
    const int lr   = lane & 15;
    const int ntn  = RES_DIM / 16;          // 256 tiles along N
    const int m0   = (wave / ntn) * 16;     // tile row in T
    const int n0   = (wave % ntn) * 16;     // tile col in RES

    const float* xrow = X    + (size_t)(m0 + lr) * IN_DIM;  // A: row M=m0+lr
    const float* wrow = W_in + (size_t)(n0 + lr) * IN_DIM;  // B: col N=n0+lr

    v8f acc = {};
    #pragma unroll 4
    for (int k = 0; k < IN_DIM; k += 4) {
        v2f a, b;
        a.x = xrow[k + 2 * half];
        a.y = xrow[k + 2 * half + 1];
        b.x = wrow[k + 2 * half];
        b.y = wrow[k + 2 * half + 1];
        acc = __builtin_amdgcn_wmma_f32_16x16x4_f32(
            /*neg_a=*/false, a, /*neg_b=*/false, b,
            /*c_mod=*/(short)0, acc, /*reuse_a=*/false, /*reuse_b=*/false);
    }

    float* urow = U + (size_t)(m0 + 8 * half) * RES_DIM + n0 + lr;
    #pragma unroll
    for (int r = 0; r < 8; ++r) urow[(size_t)r * RES_DIM] = acc[r];
}

// ---------------------------------------------------------------------------
// Async global->LDS staging helper (gfx1250 ASYNCcnt path, no VGPR roundtrip).
// GVS addressing: SGPR-pair base + per-lane 32-bit byte offset.
// VDST carries the LDS byte address (low 32 bits of generic shared pointer).
// ---------------------------------------------------------------------------
__device__ __forceinline__ void async_copy_b128(unsigned lds_addr,
                                                unsigned gbyte_off,
                                                const float* base) {
    asm volatile("global_load_async_to_lds_b128 %0, %1, %2"
                 :
                 : "v"(lds_addr), "v"(gbyte_off), "s"(base)
                 : "memory");
}
__device__ __forceinline__ void wait_asynccnt0() {
    asm volatile("s_wait_asynccnt 0" ::: "memory");
}

// ---------------------------------------------------------------------------
// Phase 2: persistent sequential scan with a grid-wide atomic barrier.
// ---------------------------------------------------------------------------
__global__ __launch_bounds__(BS)
void esn_scan(const float* __restrict__ X,
              const float* __restrict__ W_in,
              const float* __restrict__ W_res,
              const float* __restrict__ W_out,
              const float* __restrict__ U,        // may be null -> inline W_in@x
              float* __restrict__ sA,
              float* __restrict__ sB,
              unsigned int* __restrict__ cnt,
              float* __restrict__ preds) {
    __shared__ __align__(16) float s_state[RES_DIM];
    __shared__ __align__(16) float s_x[IN_DIM];

    const int tid  = threadIdx.x;
    const int bid  = blockIdx.x;
    const int rloc = tid >> 4;          // 0..15: local W_res row
    const int sub  = tid & (LPR - 1);   // 0..15: lane within row group
    const int row  = bid * ROWS_PB + rloc;

    const float4* wres4 = (const float4*)(W_res + (size_t)row * RES_DIM);
    const float4* win4  = (const float4*)(W_in  + (size_t)row * IN_DIM);
    const float*  wout  = W_out + (size_t)bid * RES_DIM;   // pred row = bid

    const unsigned lds_state = (unsigned)(uintptr_t)&s_state[0];
    const unsigned lds_x     = (unsigned)(uintptr_t)&s_x[0];

    float* cur = sA;
    float* nxt = sB;

    for (int t = 0; t <= T_STEPS; ++t) {
        // ---- prefetch this step's U row while we stage/compute ----
        if (U != nullptr && t < T_STEPS && sub == 0) {
            __builtin_prefetch(&U[(size_t)t * RES_DIM + row], 0, 0);
        }

        // ---- stage current state (result of step t-1) into LDS, async ----
        // 4096 floats = 1024 float4; 256 threads x 4 async b128 copies.
        #pragma unroll
        for (int j = 0; j < RES_DIM / (4 * BS); ++j) {      // 4 iters
            const unsigned e16 = (unsigned)(tid + j * BS) * 16u;  // byte offset
            async_copy_b128(lds_state + e16, e16, cur);
        }
        if (U == nullptr && t < T_STEPS && tid < IN_DIM / 4) {   // 64 threads
            const unsigned e16 = (unsigned)tid * 16u;
            async_copy_b128(lds_x + e16, e16, X + (size_t)t * IN_DIM);
        }
        wait_asynccnt0();
        __syncthreads();

        // ---- readout for previous step (wave 0 only) ----
        if (t > 0 && tid < 32) {
            float p = 0.0f;
            for (int k = tid; k < RES_DIM; k += 32) p += wout[k] * s_state[k];
            #pragma unroll
            for (int off = 16; off >= 1; off >>= 1) p += __shfl_xor(p, off, 32);
            if (tid == 0) preds[(size_t)(t - 1) * OUT_DIM + bid] = p;
        }
        if (t == T_STEPS) break;

        // ---- reservoir GEMV: 16 lanes per row, float4 strided ----
        float acc = 0.0f;
        const float4* s4 = (const float4*)s_state;
        #pragma unroll 4
        for (int i = 0; i < RES_DIM / (4 * LPR); ++i) {  // 64 iters
            const int idx = sub + (i << 4);
            float4 w = wres4[idx];
            float4 s = s4[idx];
            acc += w.x * s.x + w.y * s.y + w.z * s.z + w.w * s.w;
        }
        if (U == nullptr) {
            const float4* x4 = (const float4*)s_x;
            #pragma unroll
            for (int j = 0; j < IN_DIM / (4 * LPR); ++j) { // 4 iters
                const int idx = sub + (j << 4);
                float4 w = win4[idx];
                float4 xv = x4[idx];
                acc += w.x * xv.x + w.y * xv.y + w.z * xv.z + w.w * xv.w;
            }
        }
        #pragma unroll
        for (int off = 8; off >= 1; off >>= 1) acc += __shfl_xor(acc, off, 32);
        if (sub == 0) {
            float z = acc + (U ? U[(size_t)t * RES_DIM + row] : 0.0f);
            nxt[row] = tanhf(z);
        }

        // ---- grid-wide barrier (monotonic counter, device scope) ----
        __threadfence();          // publish this wave's stores device-wide
        __syncthreads();
        if (tid == 0) {
            __hip_atomic_fetch_add(cnt, 1u, __ATOMIC_RELAXED,
                                   __HIP_MEMORY_SCOPE_AGENT);
            const unsigned target = (unsigned)(t + 1) * (unsigned)NB;
            while (__hip_atomic_load(cnt, __ATOMIC_RELAXED,
                                     __HIP_MEMORY_SCOPE_AGENT) < target) {
                __builtin_amdgcn_s_sleep(2);
            }
        }
        __syncthreads();
        __threadfence();          // invalidate stale caches before reading nxt

        float* tmp = cur; cur = nxt; nxt = tmp;
    }
}

// ---------------------------------------------------------------------------
extern "C" void kernel_launch(void* const* d_in, const int* in_sizes, int n_in,
                              void* d_out, int out_size, void* d_ws, size_t ws_size,
                              hipStream_t stream) {
    const float* X      = (const float*)d_in[0];
    const float* state0 = (const float*)d_in[1];
    const float* W_in   = (const float*)d_in[2];
    const float* W_res  = (const float*)d_in[3];
    const float* W_out  = (const float*)d_in[4];
    float* preds = (float*)d_out;

    const size_t needU    = (size_t)T_STEPS * RES_DIM * sizeof(float); // 128 MB
    const size_t needRest = (size_t)2 * RES_DIM * sizeof(float) + 256;

    char* ws = (char*)d_ws;
    float* U = nullptr;
    if (ws_size >= needU + needRest) { U = (float*)ws; ws += needU; }
    float* sA = (float*)ws;               ws += (size_t)RES_DIM * sizeof(float);
    float* sB = (float*)ws;               ws += (size_t)RES_DIM * sizeof(float);
    unsigned int* cnt = (unsigned int*)ws;

    esn_init<<<(RES_DIM + 255) / 256, 256, 0, stream>>>(state0, sA, sB, cnt);

    if (U) {
        // (T/16)*(RES/16) = 131072 tiles, 1 wave each, 4 waves/block
        const int nblocks = (T_STEPS / 16) * (RES_DIM / 16) / 4; // 32768
        esn_win_gemm_wmma<<<nblocks, 128, 0, stream>>>(X, W_in, U);
    }

    esn_scan<<<NB, BS, 0, stream>>>(X, W_in, W_res, W_out, U, sA, sB, cnt, preds);
}